// TorchKernelPP_80917183857046
// MI455X (gfx1250) — compile-verified
//
#include <hip/hip_runtime.h>
#include <math.h>
#include <stdint.h>

#define T_DAYS 366
#define K_EV   64
#define W_HIST 30
#define AREA_C 100.0f
#define EPS_C  1e-5f
#define WPB    8                 // waves per block (wave32)
#define L2E    1.44269504088896340736f  // log2(e)

typedef __attribute__((ext_vector_type(2))) float v2f;
typedef __attribute__((ext_vector_type(8))) float v8f;

// One block per day d. 8 waves (256 threads, wave32).
// Computes per-day partial sums:
//   ws[d*2+0] = sum_c mask[d,c] * log(nu[d,c] + Lambda0 + EPS)
//   ws[d*2+1] = sum_c mask[d,c] * geo(d)            (without alpha*beta factor)
__global__ __launch_bounds__(32 * WPB) void hawkes_day_kernel(
    const float* __restrict__ obs,      // (T,K,3): [day, x, y]
    const uint8_t* __restrict__ mask,   // (T,K) bool bytes
    const float* __restrict__ Lambda0p,
    const float* __restrict__ alphap,
    const float* __restrict__ betap,
    const float* __restrict__ sigmap,
    float* __restrict__ ws)             // (T,2)
{
    const int d    = blockIdx.x;
    const int tid  = threadIdx.x;
    const int wv   = tid >> 5;      // wave id 0..WPB-1
    const int lane = tid & 31;
    const int n    = lane & 15;     // N index within tile
    const bool hiHalf = (lane >= 16);

    const float Lambda0 = Lambda0p[0];
    const float alpha   = alphap[0];
    const float beta    = betap[0];
    const float sigma   = sigmap[0];
    const float sig2    = sigma * sigma;
    const float s2      = 0.5f / sig2;                         // 1/(2 sigma^2)
    const float gnorm   = 1.0f / (6.283185307179586f * sig2);  // 1/(2 pi sigma^2)
    const float ab      = alpha * beta;

    __shared__ float nu_part[WPB][K_EV];
    __shared__ float red1[K_EV];
    __shared__ float red2[K_EV];

    // ---- Build A tiles from current-day points ----
    // 16x4 f32 A layout: lanes 0-15 hold (K0,K1) in (v.x,v.y); lanes 16-31 hold (K2,K3).
    // Row vector, pre-scaled by log2(e) so the WMMA output is a base-2 exponent:
    //   L2E * [2*s2*cx, 2*s2*cy, -s2*|c|^2, -1]
    v2f A[4];
#pragma unroll
    for (int mt = 0; mt < 4; ++mt) {
        const int c = mt * 16 + n;
        const float cx = obs[(d * K_EV + c) * 3 + 1];
        const float cy = obs[(d * K_EV + c) * 3 + 2];
        if (!hiHalf) {
            A[mt].x = L2E * 2.0f * s2 * cx;
            A[mt].y = L2E * 2.0f * s2 * cy;
        } else {
            A[mt].x = -L2E * s2 * (cx * cx + cy * cy);
            A[mt].y = -L2E;
        }
    }

    float acc[4][8];
#pragma unroll
    for (int mt = 0; mt < 4; ++mt)
#pragma unroll
        for (int i = 0; i < 8; ++i) acc[mt][i] = 0.0f;

    // ---- History loop: wave wv handles w = wv+1, wv+1+WPB, ... ----
    for (int w = wv + 1; w <= W_HIST; w += WPB) {
        const int h = d - w;
        if (h < 0) break;  // h only decreases as w grows
        const float wcoef = ab * gnorm * __expf(-beta * (float)w);
#pragma unroll
        for (int nt = 0; nt < 4; ++nt) {
            const int k = nt * 16 + n;
            const float hx = obs[(h * K_EV + k) * 3 + 1];
            const float hy = obs[(h * K_EV + k) * 3 + 2];
            // 4x16 f32 B layout: lanes 0-15 rows K0/K1, lanes 16-31 rows K2/K3.
            // Column vector (unscaled): [hx, hy, 1, s2*|h|^2]
            v2f B;
            if (!hiHalf) { B.x = hx;   B.y = hy; }
            else         { B.x = 1.0f; B.y = s2 * (hx * hx + hy * hy); }
            // per-N coefficient (this lane's N column of the D tile)
            const float coef = mask[h * K_EV + k] ? wcoef : 0.0f;
#pragma unroll
            for (int mt = 0; mt < 4; ++mt) {
                v8f c8 = {};
                // D[c][j] = -log2(e) * |cur_c - his_j|^2 / (2 sigma^2)
                v8f d8 = __builtin_amdgcn_wmma_f32_16x16x4_f32(
                    false, A[mt], false, B, (short)0, c8, false, false);
#pragma unroll
                for (int i = 0; i < 8; ++i)
                    acc[mt][i] += coef * __builtin_amdgcn_exp2f(d8[i]);
            }
        }
    }

    // ---- Reduce across the 16 N-lanes of each half-wave ----
    // D layout: lane L holds N=L&15; M = i (lanes 0-15) or 8+i (lanes 16-31).
#pragma unroll
    for (int mt = 0; mt < 4; ++mt) {
#pragma unroll
        for (int i = 0; i < 8; ++i) {
            float v = acc[mt][i];
            v += __shfl_xor(v, 1, 32);
            v += __shfl_xor(v, 2, 32);
            v += __shfl_xor(v, 4, 32);
            v += __shfl_xor(v, 8, 32);
            if (n == 0) {
                const int M = mt * 16 + (hiHalf ? (8 + i) : i);
                nu_part[wv][M] = v;   // unique writer per (wv, M)
            }
        }
    }
    __syncthreads();

    // ---- Per-event log term and geo term (fixed summation order) ----
    if (tid < K_EV) {
        const int c = tid;
        float nu = 0.0f;
#pragma unroll
        for (int p = 0; p < WPB; ++p) nu += nu_part[p][c];
        const float lam = nu + Lambda0;
        const uint8_t m = mask[d * K_EV + c];
        float Dv = (float)(T_DAYS - (d + 1));
        Dv = fminf(fmaxf(Dv, 0.0f), (float)W_HIST);
        const float emb = __expf(-beta);
        const float geo = emb * (1.0f - __expf(-beta * Dv)) / (1.0f - emb);
        red1[c] = m ? logf(lam + EPS_C) : 0.0f;
        red2[c] = m ? geo : 0.0f;
    }
    __syncthreads();

    if (tid == 0) {
        float s1 = 0.0f, sg = 0.0f;
        for (int c = 0; c < K_EV; ++c) { s1 += red1[c]; sg += red2[c]; }
        ws[d * 2 + 0] = s1;
        ws[d * 2 + 1] = sg;
    }
}

// Single-block deterministic reduction of the 366 per-day partials.
__global__ __launch_bounds__(128) void hawkes_final_kernel(
    const float* __restrict__ ws,
    const float* __restrict__ Lambda0p,
    const float* __restrict__ alphap,
    const float* __restrict__ betap,
    float* __restrict__ out)            // [loglik, lams1, lams2]
{
    __shared__ float r1[128];
    __shared__ float r2[128];
    const int tid = threadIdx.x;
    float s1 = 0.0f, sg = 0.0f;
    for (int dd = tid; dd < T_DAYS; dd += 128) {
        s1 += ws[dd * 2 + 0];
        sg += ws[dd * 2 + 1];
    }
    r1[tid] = s1;
    r2[tid] = sg;
    __syncthreads();
    for (int off = 64; off > 0; off >>= 1) {
        if (tid < off) { r1[tid] += r1[tid + off]; r2[tid] += r2[tid + off]; }
        __syncthreads();
    }
    if (tid == 0) {
        const float lams1 = r1[0];
        const float edo   = alphap[0] * betap[0] * r2[0];
        const float lams2 = Lambda0p[0] * AREA_C * (float)T_DAYS + edo;
        out[0] = lams1 - lams2;
        out[1] = lams1;
        out[2] = lams2;
    }
}

extern "C" void kernel_launch(void* const* d_in, const int* in_sizes, int n_in,
                              void* d_out, int out_size, void* d_ws, size_t ws_size,
                              hipStream_t stream) {
    (void)in_sizes; (void)n_in; (void)out_size; (void)ws_size;
    const float*   obs      = (const float*)d_in[0];    // (T,K,3) f32
    const uint8_t* mask     = (const uint8_t*)d_in[1];  // (T,K) bool
    const float*   Lambda0p = (const float*)d_in[2];
    const float*   alphap   = (const float*)d_in[3];
    const float*   betap    = (const float*)d_in[4];
    const float*   sigmap   = (const float*)d_in[5];
    float* ws  = (float*)d_ws;    // (T,2) floats of scratch
    float* out = (float*)d_out;   // 3 floats

    hipLaunchKernelGGL(hawkes_day_kernel, dim3(T_DAYS), dim3(32 * WPB), 0, stream,
                       obs, mask, Lambda0p, alphap, betap, sigmap, ws);
    hipLaunchKernelGGL(hawkes_final_kernel, dim3(1), dim3(128), 0, stream,
                       ws, Lambda0p, alphap, betap, out);
}